// FlowLayer_70248485094016
// MI455X (gfx1250) — compile-verified
//
#include <hip/hip_runtime.h>
#include <math.h>

// ---------------------------------------------------------------------------
// TV-L1 FlowLayer for MI455X (gfx1250, wave32).
//  * Both channel-mixing einsums run on V_WMMA_F32_16X16X4_F32 (fp32 exact).
//  * All "convs" in the reference use identically-tiled kernels -> they are
//    channel-sums of 2/3-tap stencils; p-fields are channel-uniform and are
//    stored as single-channel fields. Iteration state (~94 MB) is L2-resident.
// ---------------------------------------------------------------------------

#define Hh   56
#define Ww   56
#define HW   3136          // 56*56
#define CB   32
#define CC   64
#define TT   17
#define TM   16
#define BB   2
#define BT   34            // B*T
#define BT2  32            // B*Tm
#define NPOS1 (BT * HW)    // 106624  (bneck GEMM rows)
#define NPOS2 (BT2 * HW)   // 100352  (flow positions)
#define NFLD  (BT2 * CB * HW)   // 3,211,264 per per-channel field
#define NOUT  (BB * CC * TM * HW) // 6,422,528
#define EPSF  1e-12f
#define N_ITER 10

typedef float v8f __attribute__((ext_vector_type(8)));
typedef float v2f __attribute__((ext_vector_type(2)));

// ---- workspace layout (float offsets); total ~23.59M floats (~94.4 MB) ----
#define OFF_XB    0                         // (BT, CB, HW) bottleneck output
#define OFF_XS    (OFF_XB + BT * CB * HW)   // (BT, HW) channel-sum of xb
#define OFF_GX    (OFF_XS + BT * HW)        // per-channel fields (BT2,CB,HW)
#define OFF_GY    (OFF_GX + NFLD)
#define OFF_GRAD  (OFF_GY + NFLD)
#define OFF_RHOC  (OFF_GRAD + NFLD)
#define OFF_U1    (OFF_RHOC + NFLD)
#define OFF_U2    (OFF_U1 + NFLD)
#define OFF_U1S   (OFF_U2 + NFLD)           // (BT2,HW) channel-sum of u1
#define OFF_U2S   (OFF_U1S + NPOS2)
#define OFF_P11   (OFF_U2S + NPOS2)         // channel-uniform dual fields
#define OFF_P12   (OFF_P11 + NPOS2)
#define OFF_P21   (OFF_P12 + NPOS2)
#define OFF_P22   (OFF_P21 + NPOS2)
#define OFF_D1    (OFF_P22 + NPOS2)         // t * divergence fields
#define OFF_D2    (OFF_D1 + NPOS2)
#define OFF_MM    (OFF_D2 + NPOS2)          // 2 uints: ordered min/max
#define OFF_BNS   (OFF_MM + 2)              // 64 per-channel sums
#define OFF_BNQ   (OFF_BNS + CC)            // 64 per-channel sum-of-squares
#define OFF_BNA   (OFF_BNQ + CC)            // 64 effective BN scale
#define OFF_BNB   (OFF_BNA + CC)            // 64 effective BN bias
#define WS_TOTAL  (OFF_BNB + CC)
#define OFF_FLOW  OFF_GX                    // gx/gy dead after the loop; flow
                                            // (B,CC,TM,HW)=6,422,528 = 2*NFLD

// order-preserving float<->uint mapping for atomic min/max
__device__ __forceinline__ unsigned f2ord(float f) {
  unsigned u = __float_as_uint(f);
  return (u & 0x80000000u) ? ~u : (u | 0x80000000u);
}
__device__ __forceinline__ float ord2f(unsigned u) {
  return (u & 0x80000000u) ? __uint_as_float(u ^ 0x80000000u)
                           : __uint_as_float(~u);
}

// ---------------------------------------------------------------------------
__global__ void k_init(float* __restrict__ ws) {
  int idx = blockIdx.x * 256 + threadIdx.x;
  int zn = WS_TOTAL - OFF_U1S;
  if (idx < zn) {
    int pos = OFF_U1S + idx;
    if (pos < OFF_MM || pos >= OFF_BNS) ws[pos] = 0.0f;  // keep mm slot aside
  }
  if (idx == 0) {
    unsigned* mm = (unsigned*)(ws + OFF_MM);
    mm[0] = 0xFFFFFFFFu;  // ordered min identity
    mm[1] = 0u;           // ordered max identity
  }
}

// ---------------------------------------------------------------------------
// Bottleneck GEMM: xb[bt,o,hw] = sum_c x[b,c,t,hw] * wb[o,c]  (K=64, N=32)
// One wave -> 16 positions x 32 channels via 2x (16 WMMA f32 16x16x4).
// Also produces the global min/max via wave-reduce + 2 atomics/wave.
__global__ void k_gemm_bneck(const float* __restrict__ x,
                             const float* __restrict__ wb,
                             float* __restrict__ ws) {
  float* xb = ws + OFF_XB;
  unsigned* mm = (unsigned*)(ws + OFF_MM);
  int gwave = (blockIdx.x * 256 + threadIdx.x) >> 5;
  int lane  = threadIdx.x & 31;
  int posBase = gwave * 16;                 // HW % 16 == 0 -> tile in one frame
  int bt = posBase / HW;
  int hwBase = posBase - bt * HW;
  int b = bt / TT, t = bt - b * TT;
  int m = lane & 15;                        // A row held by this lane
  int n = lane & 15;                        // C/D column held by this lane
  int khalf = (lane >> 4) << 1;             // lanes 16..31 hold K+2,K+3
  const float* xp = x + ((size_t)(b * CC) * TT + t) * HW + hwBase + m;
  v8f acc0 = {};
  v8f acc1 = {};
  for (int kk = 0; kk < 16; ++kk) {
    int c0 = kk * 4 + khalf;
    v2f av, b0, b1;
    av.x = xp[(size_t)c0 * (TT * HW)];
    av.y = xp[(size_t)(c0 + 1) * (TT * HW)];
    b0.x = wb[n * CC + c0];          b0.y = wb[n * CC + c0 + 1];
    b1.x = wb[(n + 16) * CC + c0];   b1.y = wb[(n + 16) * CC + c0 + 1];
    acc0 = __builtin_amdgcn_wmma_f32_16x16x4_f32(false, av, false, b0,
                                                 (short)0, acc0, false, false);
    acc1 = __builtin_amdgcn_wmma_f32_16x16x4_f32(false, av, false, b1,
                                                 (short)0, acc1, false, false);
  }
  float vmax = -3.402823466e38f, vmin = 3.402823466e38f;
  int rbase = (lane >> 4) * 8;              // rows 0..7 / 8..15
  for (int r = 0; r < 8; ++r) {
    int hw = hwBase + rbase + r;
    float v0 = acc0[r], v1 = acc1[r];
    xb[((size_t)bt * CB + n) * HW + hw]        = v0;
    xb[((size_t)bt * CB + n + 16) * HW + hw]   = v1;
    vmax = fmaxf(vmax, fmaxf(v0, v1));
    vmin = fminf(vmin, fminf(v0, v1));
  }
  for (int off = 16; off > 0; off >>= 1) {
    vmax = fmaxf(vmax, __shfl_xor(vmax, off, 32));
    vmin = fminf(vmin, __shfl_xor(vmin, off, 32));
  }
  if (lane == 0) {
    atomicMin(&mm[0], f2ord(vmin));
    atomicMax(&mm[1], f2ord(vmax));
  }
}

// ---------------------------------------------------------------------------
__global__ void k_chansum(float* __restrict__ ws) {
  int idx = blockIdx.x * 256 + threadIdx.x;
  if (idx >= BT * HW) return;
  const float* xb = ws + OFF_XB;
  int bt = idx / HW, hw = idx - bt * HW;
  float s = 0.0f;
  for (int c = 0; c < CB; ++c) s += xb[((size_t)bt * CB + c) * HW + hw];
  ws[OFF_XS + idx] = s;
}

// ---------------------------------------------------------------------------
// gx/gy (channel-sum stencils, per-channel boundary overrides from x2),
// grad, rho_c = scale*(y2-x2); also zero-init u1,u2.
__global__ void k_prep(float* __restrict__ ws) {
  int idx = blockIdx.x * 256 + threadIdx.x;
  if (idx >= NPOS2) return;
  int bt2 = idx / HW, hw = idx - bt2 * HW;
  int b = bt2 >> 4, tm = bt2 & 15;
  int h = hw / Ww, w = hw - h * Ww;
  const unsigned* mm = (const unsigned*)(ws + OFF_MM);
  float mnv = ord2f(mm[0]);
  float mxv = ord2f(mm[1]);
  float sc = 255.0f / (mnv - mxv);          // note: reference uses (mn - mx)
  int btx = b * TT + tm, bty = btx + 1;
  const float* xb = ws + OFF_XB;
  const float* XsY = ws + OFF_XS + (size_t)bty * HW;
  float gxu = 0.0f, gyu = 0.0f;             // channel-uniform interior values
  if (w > 0 && w < Ww - 1) gxu = 0.5f * sc * (XsY[hw + 1] - XsY[hw - 1]);
  if (h > 0 && h < Hh - 1) gyu = 0.5f * sc * (XsY[hw + Ww] - XsY[hw - Ww]);
  float* gx = ws + OFF_GX;  float* gy = ws + OFF_GY;
  float* gr = ws + OFF_GRAD; float* rc = ws + OFF_RHOC;
  float* u1 = ws + OFF_U1;  float* u2 = ws + OFF_U2;
  for (int o = 0; o < CB; ++o) {
    size_t ix = ((size_t)btx * CB + o) * HW + hw;
    size_t iy = ((size_t)bty * CB + o) * HW + hw;
    float xv = xb[ix], yv = xb[iy];
    float g1 = gxu, g2 = gyu;
    if (w == 0)           g1 = 0.5f * sc * (xb[ix + 1] - xv);
    else if (w == Ww - 1) g1 = 0.5f * sc * (xv - xb[ix - 1]);
    if (h == 0)           g2 = 0.5f * sc * (xb[ix + Ww] - xv);
    else if (h == Hh - 1) g2 = 0.5f * sc * (xv - xb[ix - Ww]);
    size_t f = ((size_t)bt2 * CB + o) * HW + hw;
    gx[f] = g1;
    gy[f] = g2;
    gr[f] = g1 * g1 + g2 * g2 + EPSF;
    rc[f] = sc * (yv - xv);
    u1[f] = 0.0f;
    u2[f] = 0.0f;
  }
}

// ---------------------------------------------------------------------------
// D = t * 32 * (backdiff(p11)+backdiff(p12)) (channel-sum of tiled div conv)
__global__ void k_div(float* __restrict__ ws, const float* __restrict__ tp) {
  int idx = blockIdx.x * 256 + threadIdx.x;
  if (idx >= NPOS2) return;
  int bt2 = idx / HW, hw = idx - bt2 * HW;
  int h = hw / Ww;
  size_t base = (size_t)bt2 * HW + hw;
  const float* p11 = ws + OFF_P11; const float* p12 = ws + OFF_P12;
  const float* p21 = ws + OFF_P21; const float* p22 = ws + OFF_P22;
  float d1, d2;
  if (h == 0) {
    d1 = p11[base] + p12[base];
    d2 = p21[base] + p22[base];
  } else if (h < Hh - 1) {
    d1 = (p11[base] - p11[base - Ww]) + (p12[base] - p12[base - Ww]);
    d2 = (p21[base] - p21[base - Ww]) + (p22[base] - p22[base - Ww]);
  } else {
    d1 = -(p11[base - Ww] + p12[base - Ww]);
    d2 = -(p21[base - Ww] + p22[base - Ww]);
  }
  float t = *tp;
  ws[OFF_D1 + idx] = t * 32.0f * d1;
  ws[OFF_D2 + idx] = t * 32.0f * d2;
}

// ---------------------------------------------------------------------------
// per-channel u update: rho/threshold/v + t*divergence
__global__ void k_u(float* __restrict__ ws, const float* __restrict__ tp,
                    const float* __restrict__ lp) {
  int idx = blockIdx.x * 256 + threadIdx.x;   // exactly NFLD threads
  int ch = idx / HW;
  int hw = idx - ch * HW;
  int bt2 = ch >> 5;
  float t = *tp, l = *lp;
  float l_t = l * t;
  float u1v = ws[OFF_U1 + idx], u2v = ws[OFF_U2 + idx];
  float g1 = ws[OFF_GX + idx],  g2 = ws[OFF_GY + idx];
  float gr = ws[OFF_GRAD + idx], rc = ws[OFF_RHOC + idx];
  int pidx = bt2 * HW + hw;
  float D1 = ws[OFF_D1 + pidx], D2 = ws[OFF_D2 + pidx];
  float rho = rc + g1 * u1v + g2 * u2v + EPSF;
  float th = l_t * gr;
  float v1, v2;
  if (rho < -th)        { v1 = l_t * g1;  v2 = l_t * g2; }
  else if (rho > th)    { v1 = -l_t * g1; v2 = -l_t * g2; }
  else if (gr > EPSF)   { float s = -rho / gr; v1 = s * g1; v2 = s * g2; }
  else                  { v1 = 0.0f; v2 = 0.0f; }
  ws[OFF_U1 + idx] = u1v + v1 + D1;
  ws[OFF_U2 + idx] = u2v + v2 + D2;
}

// ---------------------------------------------------------------------------
__global__ void k_reduce_u(float* __restrict__ ws) {
  int idx = blockIdx.x * 256 + threadIdx.x;
  if (idx >= NPOS2) return;
  int bt2 = idx / HW, hw = idx - bt2 * HW;
  float s1 = 0.0f, s2 = 0.0f;
  for (int o = 0; o < CB; ++o) {
    size_t f = ((size_t)bt2 * CB + o) * HW + hw;
    s1 += ws[OFF_U1 + f];
    s2 += ws[OFF_U2 + f];
  }
  ws[OFF_U1S + idx] = s1;
  ws[OFF_U2S + idx] = s2;
}

// ---------------------------------------------------------------------------
// p update on channel-uniform fields (forward diff of channel-sums; the
// reference's f_grad == f_grad2, so u1x == u1y).
__global__ void k_p(float* __restrict__ ws, const float* __restrict__ tp,
                    const float* __restrict__ ap) {
  int idx = blockIdx.x * 256 + threadIdx.x;
  if (idx >= NPOS2) return;
  int bt2 = idx / HW, hw = idx - bt2 * HW;
  int h = hw / Ww;
  float t = *tp, a = *ap;
  float taut = a / t;
  float fx1 = 0.0f, fx2 = 0.0f;
  if (h < Hh - 1) {
    fx1 = ws[OFF_U1S + idx + Ww] - ws[OFF_U1S + idx];
    fx2 = ws[OFF_U2S + idx + Ww] - ws[OFF_U2S + idx];
  }
  float n1 = sqrtf(2.0f * fx1 * fx1 + EPSF);
  float n2 = sqrtf(2.0f * fx2 * fx2 + EPSF);
  float r1 = 1.0f / (1.0f + taut * n1);
  float r2 = 1.0f / (1.0f + taut * n2);
  ws[OFF_P11 + idx] = (ws[OFF_P11 + idx] + taut * fx1) * r1;
  ws[OFF_P12 + idx] = (ws[OFF_P12 + idx] + taut * fx1) * r1;
  ws[OFF_P21 + idx] = (ws[OFF_P21 + idx] + taut * fx2) * r2;
  ws[OFF_P22 + idx] = (ws[OFF_P22 + idx] + taut * fx2) * r2;
}

// ---------------------------------------------------------------------------
// Un-bottleneck GEMM: flow[b,o,tm,hw] = sum_c [u1|u2][c] * wu[o,c]  (K=N=64)
// One wave -> 16 positions x 64 channels (4 n-tiles, 16 WMMA each).
// Also accumulates per-channel sum/sumsq for BN (shfl + LDS + atomics).
__global__ void k_gemm_unbneck(const float* __restrict__ wu,
                               float* __restrict__ ws) {
  __shared__ float sSum[CC];
  __shared__ float sSq[CC];
  for (int i = threadIdx.x; i < CC; i += 256) { sSum[i] = 0.0f; sSq[i] = 0.0f; }
  __syncthreads();
  int gwave = (blockIdx.x * 256 + threadIdx.x) >> 5;
  int lane  = threadIdx.x & 31;
  int posBase = gwave * 16;
  int bt2 = posBase / HW;
  int hwBase = posBase - bt2 * HW;
  int b = bt2 >> 4, tm = bt2 & 15;
  int m = lane & 15;
  int n = lane & 15;
  int khalf = (lane >> 4) << 1;
  const float* pA1 = ws + OFF_U1 + (size_t)bt2 * CB * HW + hwBase + m;
  const float* pA2 = ws + OFF_U2 + (size_t)bt2 * CB * HW + hwBase + m;
  v8f zero = {};
  v8f acc[4] = {zero, zero, zero, zero};
  for (int kk = 0; kk < 16; ++kk) {
    int c0 = kk * 4 + khalf;
    v2f av;
    if (kk < 8) {                       // scalar-uniform branch: EXEC stays full
      av.x = pA1[(size_t)c0 * HW];
      av.y = pA1[(size_t)(c0 + 1) * HW];
    } else {
      av.x = pA2[(size_t)(c0 - 32) * HW];
      av.y = pA2[(size_t)(c0 - 31) * HW];
    }
#pragma unroll
    for (int nt = 0; nt < 4; ++nt) {
      v2f bv;
      bv.x = wu[(n + nt * 16) * CC + c0];
      bv.y = wu[(n + nt * 16) * CC + c0 + 1];
      acc[nt] = __builtin_amdgcn_wmma_f32_16x16x4_f32(false, av, false, bv,
                                                      (short)0, acc[nt],
                                                      false, false);
    }
  }
  float* flow = ws + OFF_FLOW;
  int rbase = (lane >> 4) * 8;
#pragma unroll
  for (int nt = 0; nt < 4; ++nt) {
    int o = n + nt * 16;
    float s = 0.0f, q = 0.0f;
    for (int r = 0; r < 8; ++r) {
      int hw = hwBase + rbase + r;
      float v = acc[nt][r];
      flow[(((size_t)b * CC + o) * TM + tm) * HW + hw] = v;
      s += v;
      q += v * v;
    }
    s += __shfl_down(s, 16, 32);
    q += __shfl_down(q, 16, 32);
    if (lane < 16) {
      atomicAdd(&sSum[o], s);
      atomicAdd(&sSq[o], q);
    }
  }
  __syncthreads();
  float* bns = ws + OFF_BNS;
  float* bnq = ws + OFF_BNQ;
  for (int i = threadIdx.x; i < CC; i += 256) {
    atomicAdd(&bns[i], sSum[i]);
    atomicAdd(&bnq[i], sSq[i]);
  }
}

// ---------------------------------------------------------------------------
__global__ void k_bnfin(const float* __restrict__ gamma,
                        const float* __restrict__ beta,
                        float* __restrict__ ws) {
  int o = threadIdx.x;
  if (o >= CC) return;
  float cnt = (float)NPOS2;                 // B*Tm*H*W
  float mean = ws[OFF_BNS + o] / cnt;
  float var = ws[OFF_BNQ + o] / cnt - mean * mean;
  float scl = gamma[o] * rsqrtf(var + 1e-5f);
  ws[OFF_BNA + o] = scl;
  ws[OFF_BNB + o] = beta[o] - mean * scl;
}

// ---------------------------------------------------------------------------
__global__ void k_out(const float* __restrict__ x, const float* __restrict__ ws,
                      float* __restrict__ out) {
  int idx = blockIdx.x * 256 + threadIdx.x;  // exactly NOUT threads
  int hw = idx % HW;
  int r = idx / HW;
  int tm = r % TM; r /= TM;
  int o = r % CC;
  int b = r / CC;
  float f = ws[OFF_FLOW + idx];              // same (b,o,tm,hw) layout
  float res = x[(((size_t)b * CC + o) * TT + tm) * HW + hw];
  float v = res + f * ws[OFF_BNA + o] + ws[OFF_BNB + o];
  out[idx] = fmaxf(v, 0.0f);
}

// ---------------------------------------------------------------------------
extern "C" void kernel_launch(void* const* d_in, const int* in_sizes, int n_in,
                              void* d_out, int out_size, void* d_ws,
                              size_t ws_size, hipStream_t stream) {
  (void)in_sizes; (void)n_in; (void)out_size; (void)ws_size;  // needs ~95 MB ws
  const float* x     = (const float*)d_in[0];
  const float* wb    = (const float*)d_in[1];
  const float* wu    = (const float*)d_in[2];
  const float* tp    = (const float*)d_in[9];
  const float* lp    = (const float*)d_in[10];
  const float* ap    = (const float*)d_in[11];
  const float* gamma = (const float*)d_in[12];
  const float* beta  = (const float*)d_in[13];
  float* ws  = (float*)d_ws;
  float* out = (float*)d_out;

  int zInitBlocks = (WS_TOTAL - OFF_U1S + 255) / 256;
  k_init<<<zInitBlocks, 256, 0, stream>>>(ws);
  k_gemm_bneck<<<NPOS1 / 16 / 8, 256, 0, stream>>>(x, wb, ws);   // 833 blocks
  k_chansum<<<(BT * HW + 255) / 256, 256, 0, stream>>>(ws);
  k_prep<<<NPOS2 / 256, 256, 0, stream>>>(ws);
  for (int it = 0; it < N_ITER; ++it) {
    k_div<<<NPOS2 / 256, 256, 0, stream>>>(ws, tp);
    k_u<<<NFLD / 256, 256, 0, stream>>>(ws, tp, lp);
    k_reduce_u<<<NPOS2 / 256, 256, 0, stream>>>(ws);
    k_p<<<NPOS2 / 256, 256, 0, stream>>>(ws, tp, ap);
  }
  k_gemm_unbneck<<<NPOS2 / 16 / 8, 256, 0, stream>>>(wu, ws);    // 784 blocks
  k_bnfin<<<1, 64, 0, stream>>>(gamma, beta, ws);
  k_out<<<NOUT / 256, 256, 0, stream>>>(x, ws, out);
}